// fused_uvu_TP_exp_opt_23888608101020
// MI455X (gfx1250) — compile-verified
//
#include <hip/hip_runtime.h>
#include <stdint.h>

// Fused e3nn-style uvu tensor-product + scatter for L_IN=[0,1,2], MUL=32,
// 11 CG paths, 300K edges -> 25K nodes.  (FINAL — converged instruction mix.)
//
// Design (MI455X / gfx1250):
//  - one wave32 per edge, lane = channel u (MUL=32 matches wave32 exactly)
//  - edge-uniform SH row staged via CDNA5 async global->LDS
//    (global_load_async_to_lds_b32 + s_wait_asynccnt, ASYNCcnt path)
//  - edge index / src / dst scalarized via readfirstlane -> s_load (KMcnt path)
//  - CG coefficients hardcoded (real-SH basis, e3nn ordering), ~130 FMAs/lane
//  - scatter via native global_atomic_add_f32, no-return (STOREcnt only)
//  - L2 (192 MB) residency management: weight array (422 MB, single-use) is
//    loaded non-temporally so in1 (28.8 MB, ~12x reuse) and out (112 MB,
//    atomic-accumulated ~12x) stay L2-resident; HBM traffic ~580 MB.
//  - no WMMA (deliberate): fp32-only workload; only fp32 shape is 16x16x4 and
//    the per-edge 32x9x35 contraction loses more to operand packing/permutes
//    than WMMA gains; kernel is memory/atomic-bound, not VALU-bound.

#define WAVES_PER_BLOCK 8

__device__ __forceinline__ void aadd(float* p, float v) {
  unsafeAtomicAdd(p, v);   // lowers to global_atomic_add_f32 (no return)
}

__device__ __forceinline__ float ldnt(const float* p) {
  return __builtin_nontemporal_load(p);   // global_load_b32 th:TH_LOAD_NT
}

__global__ __launch_bounds__(256)
void tp_uvu_edge_kernel(const float* __restrict__ in1,
                        const float* __restrict__ in2,
                        const float* __restrict__ wgt,
                        const int*   __restrict__ srcI,
                        const int*   __restrict__ dstI,
                        float* __restrict__ out,
                        int E)
{
  const float RS2  = 0.7071067811865476f;   // 1/sqrt(2)
  const float SQ2  = 1.4142135623730951f;   // sqrt(2)
  const float RS3  = 0.5773502691896258f;   // 1/sqrt(3)
  const float RS5  = 0.4472135954999579f;   // 1/sqrt(5)
  const float RS6  = 0.4082482904638631f;   // 1/sqrt(6)
  const float F21  = 0.7745966692414834f;   // sqrt(3/5)
  const float L222 = 1.3093073414159542f;   // sqrt(12/7)

  __shared__ float s_y[WAVES_PER_BLOCK][12];

  const int lane = threadIdx.x & 31;
  const int wv   = threadIdx.x >> 5;
  int e = __builtin_amdgcn_readfirstlane((int)(blockIdx.x * WAVES_PER_BLOCK + wv));
  if (e >= E) return;

  // ---- CDNA5 async path: stage this edge's 9 SH components into LDS ----
  if (lane < 9) {
    unsigned lds_addr = (unsigned)(uintptr_t)&s_y[wv][lane];   // low 32b = LDS byte addr
    unsigned goff     = (unsigned)((e * 9 + lane) * 4);
    uint64_t base     = (uint64_t)(uintptr_t)in2;
    asm volatile("global_load_async_to_lds_b32 %0, %1, %2"
                 :
                 : "v"(lds_addr), "v"(goff), "s"(base)
                 : "memory");
  }

  const int sN = __builtin_amdgcn_readfirstlane(srcI[e]);
  const int dN = __builtin_amdgcn_readfirstlane(dstI[e]);

  const float* __restrict__ arow = in1 + (size_t)sN * 288;
  const float* __restrict__ wrow = wgt + (size_t)e * 352;

  // gathered node features for channel u=lane; l=1/l=2 order (y,z,x)/(xy,yz,z2,xz,x2y2)
  // in1 rows are hot in L2 (regular temporal policy).
  float a0  = arow[lane];
  float a1y = arow[32 + 3*lane + 0];
  float a1z = arow[32 + 3*lane + 1];
  float a1x = arow[32 + 3*lane + 2];
  float q0  = arow[128 + 5*lane + 0];
  float q1  = arow[128 + 5*lane + 1];
  float q2  = arow[128 + 5*lane + 2];
  float q3  = arow[128 + 5*lane + 3];
  float q4  = arow[128 + 5*lane + 4];

  // single-use streaming data: non-temporal so it doesn't evict in1/out from L2
  float w0  = ldnt(wrow +   0 + lane);
  float w1  = ldnt(wrow +  32 + lane);
  float w2  = ldnt(wrow +  64 + lane);
  float w3  = ldnt(wrow +  96 + lane);
  float w4  = ldnt(wrow + 128 + lane);
  float w5  = ldnt(wrow + 160 + lane);
  float w6  = ldnt(wrow + 192 + lane);
  float w7  = ldnt(wrow + 224 + lane);
  float w8  = ldnt(wrow + 256 + lane);
  float w9  = ldnt(wrow + 288 + lane);
  float w10 = ldnt(wrow + 320 + lane);

  asm volatile("s_wait_asynccnt 0x0" ::: "memory");

  float y0  = s_y[wv][0];
  float b1y = s_y[wv][1];
  float b1z = s_y[wv][2];
  float b1x = s_y[wv][3];
  float r0  = s_y[wv][4];
  float r1  = s_y[wv][5];
  float r2  = s_y[wv][6];
  float r3  = s_y[wv][7];
  float r4  = s_y[wv][8];

  float* ob = out + (size_t)dN * 1120;

  // p0 (0x0->0), out base 0
  aadd(ob + lane, w0 * a0 * y0);
  // p1 (0x1->1), base 32
  {
    float c = w1 * a0;
    float* b = ob + 32 + 3*lane;
    aadd(b + 0, c * b1y);
    aadd(b + 1, c * b1z);
    aadd(b + 2, c * b1x);
  }
  // p2 (0x2->2), base 128
  {
    float c = w2 * a0;
    float* b = ob + 128 + 5*lane;
    aadd(b + 0, c * r0); aadd(b + 1, c * r1); aadd(b + 2, c * r2);
    aadd(b + 3, c * r3); aadd(b + 4, c * r4);
  }
  // p3 (1x0->1), base 288
  {
    float c = w3 * y0;
    float* b = ob + 288 + 3*lane;
    aadd(b + 0, c * a1y);
    aadd(b + 1, c * a1z);
    aadd(b + 2, c * a1x);
  }
  // p4 (1x1->0), base 384
  aadd(ob + 384 + lane, w4 * RS3 * (a1y*b1y + a1z*b1z + a1x*b1x));
  // p5 (1x1->2), base 416: orthonormal symmetric-traceless projection of u (x) y
  {
    float* b = ob + 416 + 5*lane;
    aadd(b + 0, w5 * RS2 * (a1x*b1y + a1y*b1x));
    aadd(b + 1, w5 * RS2 * (a1y*b1z + a1z*b1y));
    aadd(b + 2, w5 * RS6 * (2.0f*a1z*b1z - a1x*b1x - a1y*b1y));
    aadd(b + 3, w5 * RS2 * (a1x*b1z + a1z*b1x));
    aadd(b + 4, w5 * RS2 * (a1x*b1x - a1y*b1y));
  }
  // symmetric traceless matrix from y's l=2 block (wave-uniform)
  float Bxx = -r2*RS6 + r4*RS2;
  float Byy = -r2*RS6 - r4*RS2;
  float Bzz =  2.0f*r2*RS6;
  float Bxy = r0*RS2;
  float Byz = r1*RS2;
  float Bxz = r3*RS2;
  // p6 (1x2->1), base 576: sqrt(3/5) * T(y2) . a1
  {
    float c = F21 * w6;
    float* b = ob + 576 + 3*lane;
    aadd(b + 0, c * (Bxy*a1x + Byy*a1y + Byz*a1z));
    aadd(b + 1, c * (Bxz*a1x + Byz*a1y + Bzz*a1z));
    aadd(b + 2, c * (Bxx*a1x + Bxy*a1y + Bxz*a1z));
  }
  // p7 (2x0->2), base 672
  {
    float c = w7 * y0;
    float* b = ob + 672 + 5*lane;
    aadd(b + 0, c * q0); aadd(b + 1, c * q1); aadd(b + 2, c * q2);
    aadd(b + 3, c * q3); aadd(b + 4, c * q4);
  }
  // symmetric traceless matrix from u's l=2 block (per-lane)
  float Axx = -q2*RS6 + q4*RS2;
  float Ayy = -q2*RS6 - q4*RS2;
  float Azz =  2.0f*q2*RS6;
  float Axy = q0*RS2;
  float Ayz = q1*RS2;
  float Axz = q3*RS2;
  // p8 (2x1->1), base 832: sqrt(3/5) * T(u2) . y1
  {
    float c = F21 * w8;
    float* b = ob + 832 + 3*lane;
    aadd(b + 0, c * (Axy*b1x + Ayy*b1y + Ayz*b1z));
    aadd(b + 1, c * (Axz*b1x + Ayz*b1y + Azz*b1z));
    aadd(b + 2, c * (Axx*b1x + Axy*b1y + Axz*b1z));
  }
  // p9 (2x2->0), base 928
  aadd(ob + 928 + lane, w9 * RS5 * (q0*r0 + q1*r1 + q2*r2 + q3*r3 + q4*r4));
  // p10 (2x2->2), base 960: sqrt(12/7) * components(sym(A*B)); trace part
  // vanishes in the component extraction, so no explicit deviator needed.
  {
    float Sxx = Axx*Bxx + Axy*Bxy + Axz*Bxz;
    float Syy = Axy*Bxy + Ayy*Byy + Ayz*Byz;
    float Szz = Axz*Bxz + Ayz*Byz + Azz*Bzz;
    float Sxy = 0.5f*(Axx*Bxy + Axy*Byy + Axz*Byz + Bxx*Axy + Bxy*Ayy + Bxz*Ayz);
    float Syz = 0.5f*(Axy*Bxz + Ayy*Byz + Ayz*Bzz + Bxy*Axz + Byy*Ayz + Byz*Azz);
    float Sxz = 0.5f*(Axx*Bxz + Axy*Byz + Axz*Bzz + Bxx*Axz + Bxy*Ayz + Bxz*Azz);
    float c = L222 * w10;
    float* b = ob + 960 + 5*lane;
    aadd(b + 0, c * SQ2 * Sxy);
    aadd(b + 1, c * SQ2 * Syz);
    aadd(b + 2, c * RS6 * (2.0f*Szz - Sxx - Syy));
    aadd(b + 3, c * SQ2 * Sxz);
    aadd(b + 4, c * RS2 * (Sxx - Syy));
  }
}

extern "C" void kernel_launch(void* const* d_in, const int* in_sizes, int n_in,
                              void* d_out, int out_size, void* d_ws, size_t ws_size,
                              hipStream_t stream) {
  const float* in1 = (const float*)d_in[0];
  const float* in2 = (const float*)d_in[1];
  const float* w   = (const float*)d_in[2];
  const int*   src = (const int*)d_in[3];
  const int*   dst = (const int*)d_in[4];
  float* out = (float*)d_out;
  const int E = in_sizes[3];

  // edge results are accumulated with atomics -> output must start at zero
  // (harness poisons d_out; memset is graph-capturable)
  hipMemsetAsync(d_out, 0, (size_t)out_size * sizeof(float), stream);

  dim3 grid((unsigned)((E + WAVES_PER_BLOCK - 1) / WAVES_PER_BLOCK));
  tp_uvu_edge_kernel<<<grid, 256, 0, stream>>>(in1, in2, w, src, dst, out, E);
}